// GCN_pyg_64510408786077
// MI455X (gfx1250) — compile-verified
//
#include <hip/hip_runtime.h>

#define NFEAT 128

typedef float v2f  __attribute__((ext_vector_type(2)));
typedef float v8f  __attribute__((ext_vector_type(8)));

// Guaranteed-native fp32 scatter atomic: no-return form (STOREcnt, fire and
// forget), device scope so the RMW is performed at the L2 atomic units and is
// coherent across WGPs. s_endpgm's implicit wait-idle covers completion.
__device__ __forceinline__ void atomic_add_f32_dev(float* p, float v) {
    asm volatile("global_atomic_add_f32 %0, %1, off scope:SCOPE_DEV"
                 :: "v"(p), "v"(v) : "memory");
}

// ---------------- degree / normalization ----------------
__global__ void k_deg_init(float* __restrict__ deg, int n) {
    int i = blockIdx.x * blockDim.x + threadIdx.x;
    if (i < n) deg[i] = 1.0f;  // self-loop weight
}

__global__ void k_deg_accum(const int* __restrict__ col, const float* __restrict__ ew,
                            float* __restrict__ deg, int E) {
    int e = blockIdx.x * blockDim.x + threadIdx.x;
    if (e < E) atomic_add_f32_dev(&deg[col[e]], ew[e]);
}

__global__ void k_rsqrt(float* __restrict__ deg, int n) {
    int i = blockIdx.x * blockDim.x + threadIdx.x;
    if (i < n) deg[i] = rsqrtf(deg[i]);   // deg >= 1 always (self-loop)
}

// ---------------- dense GEMM: out[nrows x 128] = A[nrows x 128] * W[128 x 128] ----------------
// One wave32 computes one 16x16 output tile with V_WMMA_F32_16X16X4_F32.
// Block = (32,8): 8 waves cover 16 rows x all 128 output columns (A reuse in L1).
__global__ void __launch_bounds__(256)
k_gemm128_wmma(const float* __restrict__ A, const float* __restrict__ W,
               float* __restrict__ out, int nrows) {
    const int lane = threadIdx.x;      // 0..31
    const int wv   = threadIdx.y;      // 0..7 -> N tile
    const int m0   = blockIdx.x * 16;
    const int n0   = wv * 16;
    const int hi   = lane >> 4;        // lane half
    const int l    = lane & 15;
    if (m0 + 16 > nrows) return;       // N=100000 is a multiple of 16

    v8f acc = {};
    // A 16x4 f32 layout: lane = M, VGPR0/1 = K, lane-half selects K base (+2)
    const float* ap = A + (size_t)(m0 + l) * NFEAT + 2 * hi;
    // B 4x16 f32 layout: lane = N, VGPR0/1 = K (k, k+1), half selects K base (+2)
    const float* bp = W + (size_t)(2 * hi) * NFEAT + n0 + l;
#pragma unroll 8
    for (int k = 0; k < NFEAT; k += 4) {
        v2f a = *(const v2f*)(ap + k);
        v2f b;
        b.x = bp[(size_t)k * NFEAT];
        b.y = bp[(size_t)k * NFEAT + NFEAT];
        acc = __builtin_amdgcn_wmma_f32_16x16x4_f32(
            false, a, false, b, (short)0, acc, false, false);
    }
    // C/D 16x16 f32 layout: row = vgpr + 8*half, col = lane&15
    float* op = out + (size_t)(m0 + 8 * hi) * NFEAT + n0 + l;
#pragma unroll
    for (int v = 0; v < 8; ++v)
        op[(size_t)v * NFEAT] = acc[v];
}

// ---------------- self-loop + bias init: out = dinv[i]^2 * h[i] + b ----------------
// Also serves as the zero-init of the scatter target (ws/d_out are poisoned).
__global__ void k_selfbias(const float* __restrict__ h, const float* __restrict__ dinv,
                           const float* __restrict__ bias, float* __restrict__ out, int n) {
    int idx   = blockIdx.x * blockDim.x + threadIdx.x;
    int total = n * (NFEAT / 4);
    if (idx >= total) return;
    int node = idx >> 5;          // /(NFEAT/4)
    int f    = (idx & 31) * 4;
    float di = dinv[node];
    float s  = di * di;
    float4 hv = *(const float4*)(h + (size_t)node * NFEAT + f);
    float4 bv = *(const float4*)(bias + f);
    float4 o;
    o.x = fmaf(s, hv.x, bv.x);
    o.y = fmaf(s, hv.y, bv.y);
    o.z = fmaf(s, hv.z, bv.z);
    o.w = fmaf(s, hv.w, bv.w);
    *(float4*)(out + (size_t)node * NFEAT + f) = o;
}

// ---------------- edge scatter-add: out[col] += norm * h[row] ----------------
// One wave per edge; each lane handles 4 features (float4 gather + 4 native
// fp32 atomics). row/col/ew indices are wave-uniform -> broadcast loads.
__global__ void __launch_bounds__(256)
k_edge_agg(const int* __restrict__ row, const int* __restrict__ col,
           const float* __restrict__ ew, const float* __restrict__ dinv,
           const float* __restrict__ h, float* __restrict__ out, int E) {
    int e = blockIdx.x * blockDim.y + threadIdx.y;
    if (e >= E) return;
    int r = row[e];
    int c = col[e];
    float nrm = dinv[r] * ew[e] * dinv[c];
    int f = threadIdx.x * 4;
    float4 hv = *(const float4*)(h + (size_t)r * NFEAT + f);
    float* op = out + (size_t)c * NFEAT + f;
    atomic_add_f32_dev(op + 0, nrm * hv.x);
    atomic_add_f32_dev(op + 1, nrm * hv.y);
    atomic_add_f32_dev(op + 2, nrm * hv.z);
    atomic_add_f32_dev(op + 3, nrm * hv.w);
}

// ---------------- ReLU in place ----------------
__global__ void k_relu(float* __restrict__ p, int n4) {
    int idx = blockIdx.x * blockDim.x + threadIdx.x;
    if (idx >= n4) return;
    float4 v = *(float4*)(p + (size_t)idx * 4);
    v.x = fmaxf(v.x, 0.0f);
    v.y = fmaxf(v.y, 0.0f);
    v.z = fmaxf(v.z, 0.0f);
    v.w = fmaxf(v.w, 0.0f);
    *(float4*)(p + (size_t)idx * 4) = v;
}

extern "C" void kernel_launch(void* const* d_in, const int* in_sizes, int n_in,
                              void* d_out, int out_size, void* d_ws, size_t ws_size,
                              hipStream_t stream) {
    const float* x  = (const float*)d_in[0];
    const int*   ei = (const int*)d_in[1];
    const float* ew = (const float*)d_in[2];
    const float* W1 = (const float*)d_in[3];
    const float* b1 = (const float*)d_in[4];
    const float* W2 = (const float*)d_in[5];
    const float* b2 = (const float*)d_in[6];
    float* out = (float*)d_out;

    const int N = in_sizes[0] / NFEAT;
    const int E = in_sizes[2];
    const int* row = ei;        // edge_index[0]
    const int* col = ei + E;    // edge_index[1]

    float* ws   = (float*)d_ws;
    size_t npad = ((size_t)N + 255) & ~(size_t)255;
    float* dinv = ws;                          // N floats (deg -> dinv in place)
    float* h    = ws + npad;                   // N*128 (GEMM result)
    float* g    = h + (size_t)N * NFEAT;       // N*128 (aggregation target)

    const int T = 256;
    // --- gcn_norm ---
    k_deg_init<<<(N + T - 1) / T, T, 0, stream>>>(dinv, N);
    k_deg_accum<<<(E + T - 1) / T, T, 0, stream>>>(col, ew, dinv, E);
    k_rsqrt<<<(N + T - 1) / T, T, 0, stream>>>(dinv, N);

    dim3 wblk(32, 8);
    int mtiles = (N + 15) / 16;
    int tvec   = N * (NFEAT / 4);
    int eblk   = (E + 7) / 8;

    // --- layer 1 ---
    k_gemm128_wmma<<<mtiles, wblk, 0, stream>>>(x, W1, h, N);
    k_selfbias<<<(tvec + T - 1) / T, T, 0, stream>>>(h, dinv, b1, g, N);
    k_edge_agg<<<eblk, wblk, 0, stream>>>(row, col, ew, dinv, h, g, E);
    k_relu<<<(tvec + T - 1) / T, T, 0, stream>>>(g, tvec);

    // --- layer 2 ---
    k_gemm128_wmma<<<mtiles, wblk, 0, stream>>>(g, W2, h, N);
    k_selfbias<<<(tvec + T - 1) / T, T, 0, stream>>>(h, dinv, b2, out, N);
    k_edge_agg<<<eblk, wblk, 0, stream>>>(row, col, ew, dinv, h, out, E);
}